// LocalAggregation_34737695490779
// MI455X (gfx1250) — compile-verified
//
#include <hip/hip_runtime.h>
#include <hip/hip_bf16.h>

// ---------------------------------------------------------------------------
// PointNet++ local aggregation for MI455X (gfx1250, wave32).
// 3-pass recompute pipeline (global BatchNorm forces stats->apply ordering):
//   K0 transpose x -> xt[B][N][32]        (coalesced gathers later)
//   K1 ball query  -> idx[B][N][32]       (ascending-j scan == top-k smallest)
//   K2 gather+conv1 via V_WMMA_F32_16X16X4_F32, per-block BN1 partial sums
//   K3 finalize BN1 (scale/shift)
//   K4 gather+conv1+BN1+ReLU (LDS) -> conv2 WMMA, BN2 partial sums
//   K5 finalize BN2
//   K6 full recompute + BN2 + ReLU + max over K -> out[B][64][N]
// fp32 WMMA keeps exact precision; the op is L2-gather bound (x = 2MB/batch),
// not matrix-throughput bound, so fp32 tensor path is the right ceiling.
// ---------------------------------------------------------------------------

typedef __attribute__((ext_vector_type(2))) float v2f;
typedef __attribute__((ext_vector_type(8))) float v8f;

#define BB      4
#define NN      4096
#define CX      32
#define CIN     35
#define CINP    36
#define COUT    64
#define KS      32
#define R2      0.01f
#define EPSV    1e-5f
#define NPAIRS  (BB * (NN / 2))
#define GCONV   1024
#define CNTF    ((float)BB * (float)NN * (float)KS)

#define FSTRIDE  65   // feature tile row stride (bank-conflict-free B reads)
#define W1STRIDE 36   // W1 LDS row stride (gcd(36,64)=4 -> distinct banks)
#define W2STRIDE 68   // W2 LDS row stride (4*o + c distinct mod 64)

// ---------------------------------------------------------------- K0: x -> xt
__global__ void xpose_kernel(const float* __restrict__ x, float* __restrict__ xt) {
  __shared__ float tile[CX][65];
  int b  = blockIdx.y;
  int n0 = blockIdx.x * 64;
  int t  = threadIdx.x;
  for (int i = t; i < CX * 64; i += 256) {
    int c = i >> 6, nn = i & 63;
    tile[c][nn] = x[((size_t)b * CX + c) * NN + n0 + nn];
  }
  __syncthreads();
  for (int i = t; i < CX * 64; i += 256) {
    int nn = i >> 5, c = i & 31;
    xt[((size_t)b * NN + n0 + nn) * CX + c] = tile[c][nn];
  }
}

// ------------------------------------------------------------- K1: ball query
__global__ void ballq_kernel(const float* __restrict__ p, int* __restrict__ idx) {
  __shared__ float sx[256], sy[256], sz[256];
  int b = blockIdx.y;
  int i = blockIdx.x * 256 + threadIdx.x;
  float px = p[((size_t)b * NN + i) * 3 + 0];
  float py = p[((size_t)b * NN + i) * 3 + 1];
  float pz = p[((size_t)b * NN + i) * 3 + 2];
  int cnt = 0, first = 0;
  int* myidx = idx + ((size_t)b * NN + i) * KS;
  for (int t0 = 0; t0 < NN; t0 += 256) {
    int j = t0 + threadIdx.x;
    __syncthreads();
    sx[threadIdx.x] = p[((size_t)b * NN + j) * 3 + 0];
    sy[threadIdx.x] = p[((size_t)b * NN + j) * 3 + 1];
    sz[threadIdx.x] = p[((size_t)b * NN + j) * 3 + 2];
    __syncthreads();
    if (cnt < KS) {
      for (int jj = 0; jj < 256; ++jj) {
        float dx = sx[jj] - px, dy = sy[jj] - py, dz = sz[jj] - pz;
        float d2 = dx * dx + dy * dy + dz * dz;
        if (d2 < R2 && cnt < KS) {
          int j2 = t0 + jj;
          if (cnt == 0) first = j2;
          myidx[cnt] = j2;
          cnt++;
        }
      }
    }
  }
  for (int r = cnt; r < KS; ++r) myidx[r] = first;   // cnt >= 1 (self in ball)
}

// ------------------------------------------- gather one 36x64 feature tile
__device__ __forceinline__ void load_ftile(const float* __restrict__ p,
                                           const float* __restrict__ xt,
                                           const int* __restrict__ idxg,
                                           int b, int n0,
                                           int* ibuf, float* flds) {
  int t = threadIdx.x;
  if (t < 64) ibuf[t] = idxg[((size_t)b * NN + n0) * KS + t];
  __syncthreads();
  int col = t & 63;
  int j = ibuf[col];
  const float* xr = xt + ((size_t)b * NN + j) * CX;
  if (t < 64) {
    int nq = n0 + (col >> 5);
    const float* pj = p + ((size_t)b * NN + j) * 3;
    const float* pq = p + ((size_t)b * NN + nq) * 3;
    flds[0 * FSTRIDE + col] = pj[0] - pq[0];
    flds[1 * FSTRIDE + col] = pj[1] - pq[1];
    flds[2 * FSTRIDE + col] = pj[2] - pq[2];
    flds[35 * FSTRIDE + col] = 0.0f;               // K padding row
    #pragma unroll
    for (int c = 0; c < 16; ++c) flds[(3 + c) * FSTRIDE + col] = xr[c];
  } else {
    #pragma unroll
    for (int c = 16; c < 32; ++c) flds[(3 + c) * FSTRIDE + col] = xr[c];
  }
}

// -------- one 16x16 output tile via chained f32 WMMA (A 16x4, B 4x16, C 16x16)
__device__ __forceinline__ v8f wmma_gemm(const float* __restrict__ Wl, int wstride,
                                         const float* __restrict__ Fl,
                                         int mt, int ct, int ksteps) {
  int lane  = threadIdx.x & 31;
  int mrow  = mt * 16 + (lane & 15);
  int khalf = (lane >> 4) << 1;     // ISA: lanes 0-15 K=0,1 ; lanes 16-31 K=2,3
  int col   = ct * 16 + (lane & 15);
  v8f acc = {0.f, 0.f, 0.f, 0.f, 0.f, 0.f, 0.f, 0.f};
  for (int kk = 0; kk < ksteps; ++kk) {
    int kb = (kk << 2) + khalf;
    v2f a, bm;
    a.x  = Wl[mrow * wstride + kb];
    a.y  = Wl[mrow * wstride + kb + 1];
    bm.x = Fl[kb * FSTRIDE + col];
    bm.y = Fl[(kb + 1) * FSTRIDE + col];
    acc = __builtin_amdgcn_wmma_f32_16x16x4_f32(false, a, false, bm,
                                                (short)0, acc, false, false);
  }
  return acc;
}

// --------------------------------------------------- K2: conv1 + BN1 partials
__global__ __launch_bounds__(128) void conv1_stats_kernel(
    const float* __restrict__ p, const float* __restrict__ xt,
    const int* __restrict__ idxg, const float* __restrict__ W1,
    float* __restrict__ part1) {
  __shared__ float W1l[COUT * W1STRIDE];
  __shared__ float flds[CINP * FSTRIDE];
  __shared__ int   ibuf[64];
  int t = threadIdx.x;
  for (int i = t; i < COUT * W1STRIDE; i += 128) {
    int c = i % W1STRIDE;
    W1l[i] = (c < CIN) ? W1[(i / W1STRIDE) * CIN + c] : 0.f;
  }
  int mt = t >> 5, lane = t & 31;
  float s[8], sq[8];
  #pragma unroll
  for (int v = 0; v < 8; ++v) { s[v] = 0.f; sq[v] = 0.f; }

  for (int pair = blockIdx.x; pair < NPAIRS; pair += gridDim.x) {
    int b  = pair / (NN / 2);
    int n0 = (pair % (NN / 2)) * 2;
    __syncthreads();
    load_ftile(p, xt, idxg, b, n0, ibuf, flds);
    __syncthreads();
    #pragma unroll
    for (int ct = 0; ct < 4; ++ct) {
      v8f acc = wmma_gemm(W1l, W1STRIDE, flds, mt, ct, 9);
      #pragma unroll
      for (int v = 0; v < 8; ++v) { float y = acc[v]; s[v] += y; sq[v] += y * y; }
    }
  }
  #pragma unroll
  for (int v = 0; v < 8; ++v)
    #pragma unroll
    for (int off = 1; off < 16; off <<= 1) {
      s[v]  += __shfl_xor(s[v],  off, 32);
      sq[v] += __shfl_xor(sq[v], off, 32);
    }
  if ((lane & 15) == 0) {
    int hi = lane >> 4;
    #pragma unroll
    for (int v = 0; v < 8; ++v) {
      int o = mt * 16 + hi * 8 + v;
      part1[(size_t)blockIdx.x * 128 + o]      = s[v];
      part1[(size_t)blockIdx.x * 128 + 64 + o] = sq[v];
    }
  }
}

// ------------------------------------------------- K3/K5: BN stats -> scale/shift
__global__ void bn_finalize_kernel(const float* __restrict__ part,
                                   const float* __restrict__ g,
                                   const float* __restrict__ bta,
                                   float* __restrict__ bnp, int nblocks) {
  int o = threadIdx.x;
  if (o >= COUT) return;
  float s = 0.f, sq = 0.f;
  for (int blk = 0; blk < nblocks; ++blk) {
    s  += part[(size_t)blk * 128 + o];
    sq += part[(size_t)blk * 128 + 64 + o];
  }
  float mean  = s / CNTF;
  float var   = sq / CNTF - mean * mean;
  float scale = g[o] * rsqrtf(var + EPSV);
  bnp[o]      = scale;
  bnp[64 + o] = bta[o] - mean * scale;
}

// --------------------------------- K4: conv1+BN1+ReLU -> conv2 + BN2 partials
__global__ __launch_bounds__(128) void conv2_stats_kernel(
    const float* __restrict__ p, const float* __restrict__ xt,
    const int* __restrict__ idxg, const float* __restrict__ W1,
    const float* __restrict__ W2, const float* __restrict__ bnp1,
    float* __restrict__ part2) {
  __shared__ float W1l[COUT * W1STRIDE];
  __shared__ float W2l[COUT * W2STRIDE];
  __shared__ float flds[CINP * FSTRIDE];
  __shared__ float f1l[COUT * FSTRIDE];
  __shared__ int   ibuf[64];
  int t = threadIdx.x;
  for (int i = t; i < COUT * W1STRIDE; i += 128) {
    int c = i % W1STRIDE;
    W1l[i] = (c < CIN) ? W1[(i / W1STRIDE) * CIN + c] : 0.f;
  }
  for (int i = t; i < COUT * W2STRIDE; i += 128) {
    int c = i % W2STRIDE;
    W2l[i] = (c < COUT) ? W2[(i / W2STRIDE) * COUT + c] : 0.f;
  }
  int mt = t >> 5, lane = t & 31;
  int hi = lane >> 4, colb = lane & 15;
  float sc1[8], sh1[8];
  #pragma unroll
  for (int v = 0; v < 8; ++v) {
    int o = mt * 16 + hi * 8 + v;
    sc1[v] = bnp1[o]; sh1[v] = bnp1[64 + o];
  }
  float s[8], sq[8];
  #pragma unroll
  for (int v = 0; v < 8; ++v) { s[v] = 0.f; sq[v] = 0.f; }

  for (int pair = blockIdx.x; pair < NPAIRS; pair += gridDim.x) {
    int b  = pair / (NN / 2);
    int n0 = (pair % (NN / 2)) * 2;
    __syncthreads();
    load_ftile(p, xt, idxg, b, n0, ibuf, flds);
    __syncthreads();
    #pragma unroll
    for (int ct = 0; ct < 4; ++ct) {
      v8f acc = wmma_gemm(W1l, W1STRIDE, flds, mt, ct, 9);
      int col = ct * 16 + colb;
      #pragma unroll
      for (int v = 0; v < 8; ++v) {
        float y = acc[v] * sc1[v] + sh1[v];
        y = y > 0.f ? y : 0.f;
        f1l[(mt * 16 + hi * 8 + v) * FSTRIDE + col] = y;
      }
    }
    __syncthreads();
    #pragma unroll
    for (int ct = 0; ct < 4; ++ct) {
      v8f acc = wmma_gemm(W2l, W2STRIDE, f1l, mt, ct, 16);
      #pragma unroll
      for (int v = 0; v < 8; ++v) { float y = acc[v]; s[v] += y; sq[v] += y * y; }
    }
  }
  #pragma unroll
  for (int v = 0; v < 8; ++v)
    #pragma unroll
    for (int off = 1; off < 16; off <<= 1) {
      s[v]  += __shfl_xor(s[v],  off, 32);
      sq[v] += __shfl_xor(sq[v], off, 32);
    }
  if ((lane & 15) == 0) {
    #pragma unroll
    for (int v = 0; v < 8; ++v) {
      int o = mt * 16 + hi * 8 + v;
      part2[(size_t)blockIdx.x * 128 + o]      = s[v];
      part2[(size_t)blockIdx.x * 128 + 64 + o] = sq[v];
    }
  }
}

// --------------------------- K6: full recompute + BN2 + ReLU + max over K
__global__ __launch_bounds__(128) void final_kernel(
    const float* __restrict__ p, const float* __restrict__ xt,
    const int* __restrict__ idxg, const float* __restrict__ W1,
    const float* __restrict__ W2, const float* __restrict__ bnp1,
    const float* __restrict__ bnp2, float* __restrict__ out) {
  __shared__ float W1l[COUT * W1STRIDE];
  __shared__ float W2l[COUT * W2STRIDE];
  __shared__ float flds[CINP * FSTRIDE];
  __shared__ float f1l[COUT * FSTRIDE];
  __shared__ int   ibuf[64];
  int t = threadIdx.x;
  for (int i = t; i < COUT * W1STRIDE; i += 128) {
    int c = i % W1STRIDE;
    W1l[i] = (c < CIN) ? W1[(i / W1STRIDE) * CIN + c] : 0.f;
  }
  for (int i = t; i < COUT * W2STRIDE; i += 128) {
    int c = i % W2STRIDE;
    W2l[i] = (c < COUT) ? W2[(i / W2STRIDE) * COUT + c] : 0.f;
  }
  int mt = t >> 5, lane = t & 31;
  int hi = lane >> 4, colb = lane & 15;
  float sc1[8], sh1[8], sc2[8], sh2[8];
  #pragma unroll
  for (int v = 0; v < 8; ++v) {
    int o = mt * 16 + hi * 8 + v;
    sc1[v] = bnp1[o]; sh1[v] = bnp1[64 + o];
    sc2[v] = bnp2[o]; sh2[v] = bnp2[64 + o];
  }
  for (int pair = blockIdx.x; pair < NPAIRS; pair += gridDim.x) {
    int b  = pair / (NN / 2);
    int n0 = (pair % (NN / 2)) * 2;
    __syncthreads();
    load_ftile(p, xt, idxg, b, n0, ibuf, flds);
    __syncthreads();
    #pragma unroll
    for (int ct = 0; ct < 4; ++ct) {
      v8f acc = wmma_gemm(W1l, W1STRIDE, flds, mt, ct, 9);
      int col = ct * 16 + colb;
      #pragma unroll
      for (int v = 0; v < 8; ++v) {
        float y = acc[v] * sc1[v] + sh1[v];
        y = y > 0.f ? y : 0.f;
        f1l[(mt * 16 + hi * 8 + v) * FSTRIDE + col] = y;
      }
    }
    __syncthreads();
    float m0[8], m1[8];
    #pragma unroll
    for (int v = 0; v < 8; ++v) { m0[v] = 0.f; m1[v] = 0.f; }  // relu => max >= 0
    #pragma unroll
    for (int ct = 0; ct < 4; ++ct) {
      v8f acc = wmma_gemm(W2l, W2STRIDE, f1l, mt, ct, 16);
      #pragma unroll
      for (int v = 0; v < 8; ++v) {
        float y = acc[v] * sc2[v] + sh2[v];
        y = y > 0.f ? y : 0.f;
        if (ct < 2) m0[v] = fmaxf(m0[v], y);
        else        m1[v] = fmaxf(m1[v], y);
      }
    }
    #pragma unroll
    for (int v = 0; v < 8; ++v)
      #pragma unroll
      for (int off = 1; off < 16; off <<= 1) {
        m0[v] = fmaxf(m0[v], __shfl_xor(m0[v], off, 32));
        m1[v] = fmaxf(m1[v], __shfl_xor(m1[v], off, 32));
      }
    if ((lane & 15) == 0) {
      #pragma unroll
      for (int v = 0; v < 8; ++v) {
        int o = mt * 16 + hi * 8 + v;
        out[((size_t)b * COUT + o) * NN + n0]     = m0[v];
        out[((size_t)b * COUT + o) * NN + n0 + 1] = m1[v];
      }
    }
  }
}

// ---------------------------------------------------------------------------
extern "C" void kernel_launch(void* const* d_in, const int* in_sizes, int n_in,
                              void* d_out, int out_size, void* d_ws, size_t ws_size,
                              hipStream_t stream) {
  const float* p  = (const float*)d_in[0];
  const float* x  = (const float*)d_in[1];
  const float* W1 = (const float*)d_in[2];
  const float* g1 = (const float*)d_in[3];
  const float* b1 = (const float*)d_in[4];
  const float* W2 = (const float*)d_in[5];
  const float* g2 = (const float*)d_in[6];
  const float* b2 = (const float*)d_in[7];
  float* out = (float*)d_out;

  char*  ws    = (char*)d_ws;
  int*   idx   = (int*)ws;                                        // 2 MB
  float* xt    = (float*)(ws + (size_t)BB * NN * KS * 4);         // 2 MB
  float* part1 = (float*)(ws + (size_t)BB * NN * KS * 4
                             + (size_t)BB * NN * CX * 4);         // 512 KB
  float* part2 = part1 + (size_t)GCONV * 128;                     // 512 KB
  float* bnp1  = part2 + (size_t)GCONV * 128;                     // 512 B
  float* bnp2  = bnp1 + 128;                                      // 512 B

  xpose_kernel<<<dim3(NN / 64, BB), 256, 0, stream>>>(x, xt);
  ballq_kernel<<<dim3(NN / 256, BB), 256, 0, stream>>>(p, idx);
  conv1_stats_kernel<<<GCONV, 128, 0, stream>>>(p, xt, idx, W1, part1);
  bn_finalize_kernel<<<1, 64, 0, stream>>>(part1, g1, b1, bnp1, GCONV);
  conv2_stats_kernel<<<GCONV, 128, 0, stream>>>(p, xt, idx, W1, W2, bnp1, part2);
  bn_finalize_kernel<<<1, 64, 0, stream>>>(part2, g2, b2, bnp2, GCONV);
  final_kernel<<<GCONV, 128, 0, stream>>>(p, xt, idx, W1, W2, bnp1, bnp2, out);
}